// AFNO2D_62766652063970
// MI455X (gfx1250) — compile-verified
//
#include <hip/hip_runtime.h>
#include <hip/hip_bf16.h>

// ---------------------------------------------------------------------------
// AFNO2D for MI455X (gfx1250): rfft2 -> block-diag complex MLP (WMMA fp32)
//                              -> irfft2 -> + b1 + x
// Shapes: x (4,768,128,256) f32, w1/w2 (8,96,96,2) f32, b1 (768) f32.
// Spectrum stored as separate re/im planes of size B*C*H*Wf in d_ws.
// FFT normalization (ortho fwd+inv = 1/(H*W)) folded into the last kernel;
// legal because ReLU is positively homogeneous and softshrink(.,0)=identity.
//
// MLP kernel uses M-tile = 32 frequency positions per workgroup (2 C-tiles
// per wave) so each weight (B) fragment feeds 2 WMMAs -> halves L2 weight
// traffic vs a 16-position tile and doubles wmma per vmem load.
// ---------------------------------------------------------------------------

#define B_   4
#define C_   768
#define H_   128
#define W_   256
#define WF_  129
#define NB_  8
#define BS_  96
#define P_   (H_ * WF_)          // 16512 frequency positions per (b,c)
#define MT_  32                  // positions per workgroup (M tile)
#define PTILES_ (P_ / MT_)       // 516 (exact)

typedef float v2f __attribute__((ext_vector_type(2)));
typedef float v8f __attribute__((ext_vector_type(8)));

__device__ __forceinline__ unsigned bitrev(unsigned x, int bits) {
    unsigned r = 0;
#pragma unroll
    for (int i = 0; i < 8; ++i) {
        if (i < bits) { r = (r << 1) | (x & 1u); x >>= 1; }
    }
    return r;
}

// ---------------------------------------------------------------------------
// Kernel 1: forward 256-pt real FFT along W for every row (B*C*H rows).
// Unnormalized. Writes Wf=129 bins to split re/im planes (row-major).
// ---------------------------------------------------------------------------
__global__ __launch_bounds__(128) void fft_row_fwd(const float* __restrict__ x,
                                                   float* __restrict__ sre,
                                                   float* __restrict__ sim) {
    __shared__ float re[W_], im[W_];
    const size_t row = blockIdx.x;               // (b*768+c)*128 + h
    const float* xr = x + row * W_;
    const int t = threadIdx.x;

#pragma unroll
    for (int j = t; j < W_; j += 128) {
        unsigned br = bitrev((unsigned)j, 8);
        re[br] = xr[j];
        im[br] = 0.0f;
    }
#pragma unroll
    for (int s = 1; s <= 8; ++s) {
        __syncthreads();
        const int m = 1 << s, half = m >> 1;
        const int grp = t / half, j = t % half;
        const int i1 = grp * m + j, i2 = i1 + half;
        const float ang = -6.28318530717958647692f * (float)j / (float)m;
        const float wr = __cosf(ang), wi = __sinf(ang);
        const float vr = re[i2], vi = im[i2];
        const float tr = vr * wr - vi * wi;
        const float ti = vr * wi + vi * wr;
        const float ur = re[i1], ui = im[i1];
        re[i1] = ur + tr; im[i1] = ui + ti;
        re[i2] = ur - tr; im[i2] = ui - ti;
    }
    __syncthreads();
    const size_t obase = row * WF_;
    sre[obase + t] = re[t];
    sim[obase + t] = im[t];
    if (t == 0) { sre[obase + 128] = re[128]; sim[obase + 128] = im[128]; }
}

// ---------------------------------------------------------------------------
// Kernel 2/4: in-place 128-pt complex FFT along H (stride WF_), sign = -1/+1.
// One block per column (B*C*Wf columns), 64 threads. Unnormalized.
// ---------------------------------------------------------------------------
__global__ __launch_bounds__(64) void fft_col(float* __restrict__ sre,
                                              float* __restrict__ sim,
                                              float sign) {
    __shared__ float re[H_], im[H_];
    const size_t col = blockIdx.x;               // bc * 129 + wf
    const size_t bc = col / WF_;
    const int wf = (int)(col % WF_);
    const size_t base = bc * (size_t)P_ + (size_t)wf;
    const int t = threadIdx.x;

#pragma unroll
    for (int h = t; h < H_; h += 64) {
        unsigned br = bitrev((unsigned)h, 7);
        re[br] = sre[base + (size_t)h * WF_];
        im[br] = sim[base + (size_t)h * WF_];
    }
#pragma unroll
    for (int s = 1; s <= 7; ++s) {
        __syncthreads();
        const int m = 1 << s, half = m >> 1;
        const int grp = t / half, j = t % half;
        const int i1 = grp * m + j, i2 = i1 + half;
        const float ang = sign * 6.28318530717958647692f * (float)j / (float)m;
        const float wr = __cosf(ang), wi = __sinf(ang);
        const float vr = re[i2], vi = im[i2];
        const float tr = vr * wr - vi * wi;
        const float ti = vr * wi + vi * wr;
        const float ur = re[i1], ui = im[i1];
        re[i1] = ur + tr; im[i1] = ui + ti;
        re[i2] = ur - tr; im[i2] = ui - ti;
    }
    __syncthreads();
#pragma unroll
    for (int h = t; h < H_; h += 64) {
        sre[base + (size_t)h * WF_] = re[h];
        sim[base + (size_t)h * WF_] = im[h];
    }
}

// ---------------------------------------------------------------------------
// Kernel 3: block-diagonal complex MLP at every frequency position.
//   o = relu_c(X @ W1c) @ W2c   (per block k, 96->96->96, complex)
// WMMA mapping: M = positions (2 tiles of 16), N = 16 outputs, K = inputs
// (step 4, fp32). 192 threads = 6 waves; wave w owns output tile w (96=6*16).
// A fragment (16x4 f32): lane<16 -> M=lane, K={k0,k0+1}; lane>=16 -> K={k0+2,k0+3}.
// B fragment (4x16 f32): N=lane%16, same K split across lane halves / vgprs.
// C/D (16x16 f32): vgpr r -> M = r + (lane<16 ? 0 : 8), N = lane%16.
// Each B fragment is reused by both M tiles (halves weight traffic).
// ---------------------------------------------------------------------------
__global__ __launch_bounds__(192) void afno_block_mlp(float* __restrict__ sre,
                                                      float* __restrict__ sim,
                                                      const float* __restrict__ w1,
                                                      const float* __restrict__ w2) {
    __shared__ float Xre[BS_ * MT_], Xim[BS_ * MT_];
    __shared__ float Hre[BS_ * MT_], Him[BS_ * MT_];

    const int ptile = blockIdx.x;                // 0..515
    const int k     = blockIdx.y;                // 0..7
    const int b     = blockIdx.z;                // 0..3
    const int pos0  = ptile * MT_;
    const int tid   = threadIdx.x;

    float* xbase_re = sre + ((size_t)(b * C_ + k * BS_)) * P_ + pos0;
    float* xbase_im = sim + ((size_t)(b * C_ + k * BS_)) * P_ + pos0;

    // Cooperative stage of X block (96 channels x 32 positions) into LDS,
    // layout [i][m] so A-fragment loads are conflict-free.
    for (int idx = tid; idx < BS_ * MT_; idx += 192) {
        const int i = idx >> 5, m = idx & 31;
        Xre[idx] = xbase_re[(size_t)i * P_ + m];
        Xim[idx] = xbase_im[(size_t)i * P_ + m];
    }
    __syncthreads();

    const int wave  = tid >> 5;                  // 0..5 -> output n-tile
    const int lane  = tid & 31;
    const int mrow  = lane & 15;                 // A-fragment M row (tile 0)
    const int khalf = (lane < 16) ? 0 : 2;       // K sub-offset for this half
    const int ocol  = wave * 16 + (lane & 15);   // output channel in block
    const int mD    = (lane < 16) ? 0 : 8;       // C/D M offset

    const float* w1k = w1 + (size_t)k * BS_ * BS_ * 2;   // [i][o][2]
    const float* w2k = w2 + (size_t)k * BS_ * BS_ * 2;

    // ---------------- Layer 1: X @ W1c ----------------
    v8f rr0 = {}, ii0 = {}, ri0 = {}, ir0 = {};
    v8f rr1 = {}, ii1 = {}, ri1 = {}, ir1 = {};
#pragma unroll 4
    for (int kk = 0; kk < BS_ / 4; ++kk) {
        const int i0 = kk * 4 + khalf;
        v2f ar0, ai0, ar1, ai1, br, bi;
        ar0.x = Xre[i0 * MT_ + mrow];        ar0.y = Xre[(i0 + 1) * MT_ + mrow];
        ai0.x = Xim[i0 * MT_ + mrow];        ai0.y = Xim[(i0 + 1) * MT_ + mrow];
        ar1.x = Xre[i0 * MT_ + mrow + 16];   ar1.y = Xre[(i0 + 1) * MT_ + mrow + 16];
        ai1.x = Xim[i0 * MT_ + mrow + 16];   ai1.y = Xim[(i0 + 1) * MT_ + mrow + 16];
        br.x = w1k[(i0 * BS_ + ocol) * 2 + 0];
        br.y = w1k[((i0 + 1) * BS_ + ocol) * 2 + 0];
        bi.x = w1k[(i0 * BS_ + ocol) * 2 + 1];
        bi.y = w1k[((i0 + 1) * BS_ + ocol) * 2 + 1];
        rr0 = __builtin_amdgcn_wmma_f32_16x16x4_f32(false, ar0, false, br, (short)0, rr0, false, false);
        ii0 = __builtin_amdgcn_wmma_f32_16x16x4_f32(false, ai0, false, bi, (short)0, ii0, false, false);
        ri0 = __builtin_amdgcn_wmma_f32_16x16x4_f32(false, ar0, false, bi, (short)0, ri0, false, false);
        ir0 = __builtin_amdgcn_wmma_f32_16x16x4_f32(false, ai0, false, br, (short)0, ir0, false, false);
        rr1 = __builtin_amdgcn_wmma_f32_16x16x4_f32(false, ar1, false, br, (short)0, rr1, false, false);
        ii1 = __builtin_amdgcn_wmma_f32_16x16x4_f32(false, ai1, false, bi, (short)0, ii1, false, false);
        ri1 = __builtin_amdgcn_wmma_f32_16x16x4_f32(false, ar1, false, bi, (short)0, ri1, false, false);
        ir1 = __builtin_amdgcn_wmma_f32_16x16x4_f32(false, ai1, false, br, (short)0, ir1, false, false);
    }

    // ComplexReLU (cartesian), D-layout -> LDS [o][m] so layer-2 A loads work.
#pragma unroll
    for (int r = 0; r < 8; ++r) {
        Hre[ocol * MT_ + (r + mD)]      = fmaxf(rr0[r] - ii0[r], 0.0f);
        Him[ocol * MT_ + (r + mD)]      = fmaxf(ri0[r] + ir0[r], 0.0f);
        Hre[ocol * MT_ + (r + mD + 16)] = fmaxf(rr1[r] - ii1[r], 0.0f);
        Him[ocol * MT_ + (r + mD + 16)] = fmaxf(ri1[r] + ir1[r], 0.0f);
    }
    __syncthreads();

    // ---------------- Layer 2: H @ W2c ----------------
    v8f rr0b = {}, ii0b = {}, ri0b = {}, ir0b = {};
    v8f rr1b = {}, ii1b = {}, ri1b = {}, ir1b = {};
#pragma unroll 4
    for (int kk = 0; kk < BS_ / 4; ++kk) {
        const int i0 = kk * 4 + khalf;
        v2f ar0, ai0, ar1, ai1, br, bi;
        ar0.x = Hre[i0 * MT_ + mrow];        ar0.y = Hre[(i0 + 1) * MT_ + mrow];
        ai0.x = Him[i0 * MT_ + mrow];        ai0.y = Him[(i0 + 1) * MT_ + mrow];
        ar1.x = Hre[i0 * MT_ + mrow + 16];   ar1.y = Hre[(i0 + 1) * MT_ + mrow + 16];
        ai1.x = Him[i0 * MT_ + mrow + 16];   ai1.y = Him[(i0 + 1) * MT_ + mrow + 16];
        br.x = w2k[(i0 * BS_ + ocol) * 2 + 0];
        br.y = w2k[((i0 + 1) * BS_ + ocol) * 2 + 0];
        bi.x = w2k[(i0 * BS_ + ocol) * 2 + 1];
        bi.y = w2k[((i0 + 1) * BS_ + ocol) * 2 + 1];
        rr0b = __builtin_amdgcn_wmma_f32_16x16x4_f32(false, ar0, false, br, (short)0, rr0b, false, false);
        ii0b = __builtin_amdgcn_wmma_f32_16x16x4_f32(false, ai0, false, bi, (short)0, ii0b, false, false);
        ri0b = __builtin_amdgcn_wmma_f32_16x16x4_f32(false, ar0, false, bi, (short)0, ri0b, false, false);
        ir0b = __builtin_amdgcn_wmma_f32_16x16x4_f32(false, ai0, false, br, (short)0, ir0b, false, false);
        rr1b = __builtin_amdgcn_wmma_f32_16x16x4_f32(false, ar1, false, br, (short)0, rr1b, false, false);
        ii1b = __builtin_amdgcn_wmma_f32_16x16x4_f32(false, ai1, false, bi, (short)0, ii1b, false, false);
        ri1b = __builtin_amdgcn_wmma_f32_16x16x4_f32(false, ar1, false, bi, (short)0, ri1b, false, false);
        ir1b = __builtin_amdgcn_wmma_f32_16x16x4_f32(false, ai1, false, br, (short)0, ir1b, false, false);
    }

    // softshrink(lambda=0) == identity; write back in place (disjoint region,
    // all reads of this region happened before this point within the block).
#pragma unroll
    for (int r = 0; r < 8; ++r) {
        const int pos = r + mD;
        xbase_re[(size_t)ocol * P_ + pos]      = rr0b[r] - ii0b[r];
        xbase_im[(size_t)ocol * P_ + pos]      = ri0b[r] + ir0b[r];
        xbase_re[(size_t)ocol * P_ + pos + 16] = rr1b[r] - ii1b[r];
        xbase_im[(size_t)ocol * P_ + pos + 16] = ri1b[r] + ir1b[r];
    }
}

// ---------------------------------------------------------------------------
// Kernel 5: inverse 256-pt real FFT along W (Hermitian extension), then
// out = y / (H*W) + b1[c] + x.
// ---------------------------------------------------------------------------
__global__ __launch_bounds__(128) void ifft_row_add(const float* __restrict__ sre,
                                                    const float* __restrict__ sim,
                                                    const float* __restrict__ x,
                                                    const float* __restrict__ b1,
                                                    float* __restrict__ out) {
    __shared__ float re[W_], im[W_];
    const size_t row = blockIdx.x;
    const size_t sbase = row * WF_;
    const int t = threadIdx.x;

#pragma unroll
    for (int j = t; j < W_; j += 128) {
        float vr, vi;
        if (j <= 128) { vr = sre[sbase + j];        vi =  sim[sbase + j]; }
        else          { vr = sre[sbase + (W_ - j)]; vi = -sim[sbase + (W_ - j)]; }
        unsigned br = bitrev((unsigned)j, 8);
        re[br] = vr;
        im[br] = vi;
    }
#pragma unroll
    for (int s = 1; s <= 8; ++s) {
        __syncthreads();
        const int m = 1 << s, half = m >> 1;
        const int grp = t / half, j = t % half;
        const int i1 = grp * m + j, i2 = i1 + half;
        const float ang = 6.28318530717958647692f * (float)j / (float)m;  // inverse
        const float wr = __cosf(ang), wi = __sinf(ang);
        const float vr = re[i2], vi = im[i2];
        const float tr = vr * wr - vi * wi;
        const float ti = vr * wi + vi * wr;
        const float ur = re[i1], ui = im[i1];
        re[i1] = ur + tr; im[i1] = ui + ti;
        re[i2] = ur - tr; im[i2] = ui - ti;
    }
    __syncthreads();

    const size_t c = (row / H_) % C_;
    const float bias = b1[c];
    const float scale = 1.0f / (float)(H_ * W_);   // fold both ortho norms
    const float* xr = x + row * W_;
    float* orow = out + row * W_;
#pragma unroll
    for (int j = t; j < W_; j += 128) {
        orow[j] = re[j] * scale + bias + xr[j];
    }
}

// ---------------------------------------------------------------------------
extern "C" void kernel_launch(void* const* d_in, const int* in_sizes, int n_in,
                              void* d_out, int out_size, void* d_ws, size_t ws_size,
                              hipStream_t stream) {
    const float* x  = (const float*)d_in[0];   // (4,768,128,256)
    const float* w1 = (const float*)d_in[1];   // (8,96,96,2)
    const float* b1 = (const float*)d_in[2];   // (1,768,1,1)
    const float* w2 = (const float*)d_in[3];   // (8,96,96,2)
    float* out = (float*)d_out;

    const size_t nspec = (size_t)B_ * C_ * H_ * WF_;   // 50,724,864 floats/plane
    float* sre = (float*)d_ws;
    float* sim = sre + nspec;
    (void)in_sizes; (void)n_in; (void)out_size; (void)ws_size;

    const int nrows = B_ * C_ * H_;    // 393216
    const int ncols = B_ * C_ * WF_;   // 396288

    fft_row_fwd<<<nrows, 128, 0, stream>>>(x, sre, sim);
    fft_col<<<ncols, 64, 0, stream>>>(sre, sim, -1.0f);

    dim3 mlp_grid(PTILES_, NB_, B_);   // (516, 8, 4)
    afno_block_mlp<<<mlp_grid, 192, 0, stream>>>(sre, sim, w1, w2);

    fft_col<<<ncols, 64, 0, stream>>>(sre, sim, 1.0f);
    ifft_row_add<<<nrows, 128, 0, stream>>>(sre, sim, x, b1, out);
}